// GraphSAGEEncoder_22617297780858
// MI455X (gfx1250) — compile-verified
//
#include <hip/hip_runtime.h>
#include <hip/hip_bf16.h>

#define N_NODES 50000
#define N_EDGES 800000
#define IN_F 10
#define HID 256
#define EMB 128
#define M_PAD 50048   // 391 * 128, covers N_NODES
#define BN_EPS 1e-5f
#define BN_ROWS 256
#define EPB 16        // edges per block in scatter

typedef __attribute__((ext_vector_type(2))) float v2f;
typedef __attribute__((ext_vector_type(8))) float v8f;

__global__ void zero_f32(float* __restrict__ p, int n) {
    int i = blockIdx.x * blockDim.x + threadIdx.x;
    if (i < n) p[i] = 0.0f;
}

// tmp[row, n] = x[row,:] @ w[:, n] + b[n]   (K = 10, not WMMA-worthy)
__global__ void input_proj(const float* __restrict__ x, const float* __restrict__ w,
                           const float* __restrict__ b, float* __restrict__ tmp) {
    int row = blockIdx.x;
    int n = threadIdx.x;
    float acc = b[n];
#pragma unroll
    for (int k = 0; k < IN_F; ++k)
        acc = fmaf(x[(size_t)row * IN_F + k], w[k * HID + n], acc);
    tmp[(size_t)row * HID + n] = acc;
}

// Column-wise partial sums -> f64 atomics (hardware global_atomic_add_f64)
__global__ void bn_stats(const float* __restrict__ tmp,
                         double* __restrict__ sum, double* __restrict__ ssq) {
    int n = threadIdx.x;
    int r0 = blockIdx.x * BN_ROWS;
    double s = 0.0, ss = 0.0;
    for (int i = 0; i < BN_ROWS; ++i) {
        int r = r0 + i;
        if (r >= N_NODES) break;
        float v = tmp[(size_t)r * HID + n];
        s += v;
        ss += (double)v * (double)v;
    }
    unsafeAtomicAdd(&sum[n], s);
    unsafeAtomicAdd(&ssq[n], ss);
}

__global__ void bn_finalize(const double* __restrict__ sum, const double* __restrict__ ssq,
                            float* __restrict__ mu, float* __restrict__ rs) {
    int n = threadIdx.x;
    double m = sum[n] * (1.0 / (double)N_NODES);
    double v = ssq[n] * (1.0 / (double)N_NODES) - m * m;
    mu[n] = (float)m;
    rs[n] = rsqrtf((float)v + BN_EPS);
}

__global__ void bn_apply_relu(const float* __restrict__ tmp, const float* __restrict__ mu,
                              const float* __restrict__ rs, const float* __restrict__ g,
                              const float* __restrict__ be, float* __restrict__ out) {
    int n = threadIdx.x;
    int r = blockIdx.x;
    float v = tmp[(size_t)r * HID + n];
    float o = (v - mu[n]) * rs[n] * g[n] + be[n];
    out[(size_t)r * HID + n] = fmaxf(o, 0.0f);
}

// One 256-wide feature row per edge; atomic adds land in L2 (agg fits easily).
__global__ void sage_scatter(const float* __restrict__ h, const int* __restrict__ ei,
                             float* __restrict__ agg, float* __restrict__ cnt) {
    int e0 = blockIdx.x * EPB;
    int n = threadIdx.x;
    for (int i = 0; i < EPB; ++i) {
        int e = e0 + i;
        if (e >= N_EDGES) return;
        int src = ei[e];
        int dst = ei[N_EDGES + e];
        float v = h[(size_t)src * HID + n];
        unsafeAtomicAdd(&agg[(size_t)dst * HID + n], v);
        if (n == 0) unsafeAtomicAdd(&cnt[dst], 1.0f);
    }
}

__global__ void sage_divide(float* __restrict__ agg, const float* __restrict__ cnt) {
    int r = blockIdx.x;
    int n = threadIdx.x;
    float inv = 1.0f / fmaxf(cnt[r], 1.0f);
    agg[(size_t)r * HID + n] *= inv;
}

// C[M,N] = A0[M,256] @ B0[256,N] + A1[M,256] @ B1[256,N] + bias[N]
// 4 waves/block; each wave owns a 32x64 output slab (2x4 v8f accumulators),
// so every B fragment is reused by two WMMAs (8 WMMA per 2 b64 + 8 b32 loads).
// fp32 WMMA layouts: A 16x4 (lane&15 = M, lane>>4 selects K-pair), B 4x16
// (lane&15 = N), C/D 16x16 f32 in 8 VGPRs (VGPR j -> M = 8*(lane>>4) + j).
__global__ void __launch_bounds__(128)
gemm_dual_wmma(const float* __restrict__ A0, const float* __restrict__ B0,
               const float* __restrict__ A1, const float* __restrict__ B1,
               const float* __restrict__ bias, float* __restrict__ C,
               int N, int Mstore) {
    const int lane = threadIdx.x & 31;
    const int wave = threadIdx.x >> 5;
    const int half = lane >> 4;   // selects K pair within the 16x16x4 op
    const int mm   = lane & 15;   // M index (A) / N index (B, C/D)
    const int m0 = blockIdx.x * 128 + wave * 32;
    const int n0 = blockIdx.y * 64;

    v8f acc[2][4] = {};

#pragma unroll
    for (int pass = 0; pass < 2; ++pass) {
        const float* __restrict__ A = pass ? A1 : A0;
        const float* __restrict__ B = pass ? B1 : B0;
        const float* arow0 = A + (size_t)(m0 + mm) * HID + 2 * half;
        const float* arow1 = arow0 + (size_t)16 * HID;
        for (int k = 0; k < HID; k += 4) {
            v2f a0, a1;
            a0.x = arow0[k];
            a0.y = arow0[k + 1];             // merged b64 load
            a1.x = arow1[k];
            a1.y = arow1[k + 1];             // merged b64 load
            const float* bk = B + (size_t)(k + 2 * half) * N + n0 + mm;
#pragma unroll
            for (int t = 0; t < 4; ++t) {
                v2f b;
                b.x = bk[t * 16];
                b.y = bk[t * 16 + N];
                acc[0][t] = __builtin_amdgcn_wmma_f32_16x16x4_f32(
                    false, a0, false, b, (short)0, acc[0][t], false, false);
                acc[1][t] = __builtin_amdgcn_wmma_f32_16x16x4_f32(
                    false, a1, false, b, (short)0, acc[1][t], false, false);
            }
        }
    }

#pragma unroll
    for (int g = 0; g < 2; ++g) {
#pragma unroll
        for (int t = 0; t < 4; ++t) {
            int n = n0 + t * 16 + mm;
            float bi = bias[n];
#pragma unroll
            for (int j = 0; j < 8; ++j) {
                int m = m0 + g * 16 + 8 * half + j;
                if (m < Mstore)
                    C[(size_t)m * N + n] = acc[g][t][j] + bi;
            }
        }
    }
}

extern "C" void kernel_launch(void* const* d_in, const int* in_sizes, int n_in,
                              void* d_out, int out_size, void* d_ws, size_t ws_size,
                              hipStream_t stream) {
    const float* x     = (const float*)d_in[0];
    const int*   ei    = (const int*)d_in[1];
    const float* w_in  = (const float*)d_in[2];
    const float* b_in  = (const float*)d_in[3];
    const float* g_in  = (const float*)d_in[4];
    const float* be_in = (const float*)d_in[5];
    const float* WL[3] = {(const float*)d_in[6],  (const float*)d_in[11], (const float*)d_in[16]};
    const float* BLb[3]= {(const float*)d_in[7],  (const float*)d_in[12], (const float*)d_in[17]};
    const float* WR[3] = {(const float*)d_in[8],  (const float*)d_in[13], (const float*)d_in[18]};
    const float* G[2]  = {(const float*)d_in[9],  (const float*)d_in[14]};
    const float* BE[2] = {(const float*)d_in[10], (const float*)d_in[15]};
    float* out = (float*)d_out;

    const size_t bigN = (size_t)M_PAD * HID;          // 12,812,288 floats
    float*  buf_h   = (float*)d_ws;
    float*  buf_tmp = buf_h + bigN;
    float*  buf_agg = buf_tmp + bigN;
    float*  cnt     = buf_agg + bigN;
    double* sum     = (double*)(cnt + M_PAD);          // 8B-aligned (offset is even)
    double* ssq     = sum + HID;
    float*  mu      = (float*)(ssq + HID);
    float*  rs      = mu + HID;

    dim3 b256(256);

    // Stage 0: input projection + BN + ReLU -> buf_h
    input_proj<<<N_NODES, b256, 0, stream>>>(x, w_in, b_in, buf_tmp);
    zero_f32<<<(1024 + 255) / 256, b256, 0, stream>>>((float*)sum, 1024); // 512 doubles
    bn_stats<<<(N_NODES + BN_ROWS - 1) / BN_ROWS, b256, 0, stream>>>(buf_tmp, sum, ssq);
    bn_finalize<<<1, b256, 0, stream>>>(sum, ssq, mu, rs);
    bn_apply_relu<<<N_NODES, b256, 0, stream>>>(buf_tmp, mu, rs, g_in, be_in, buf_h);

    for (int L = 0; L < 3; ++L) {
        const int N = (L == 2) ? EMB : HID;
        zero_f32<<<(int)((bigN + 255) / 256), b256, 0, stream>>>(buf_agg, (int)bigN);
        zero_f32<<<(M_PAD + 255) / 256, b256, 0, stream>>>(cnt, M_PAD);
        sage_scatter<<<(N_EDGES + EPB - 1) / EPB, b256, 0, stream>>>(buf_h, ei, buf_agg, cnt);
        sage_divide<<<N_NODES, b256, 0, stream>>>(buf_agg, cnt);

        float* target = (L == 2) ? out : buf_tmp;
        int mstore    = (L == 2) ? N_NODES : M_PAD;
        dim3 grid(M_PAD / 128, N / 64);
        gemm_dual_wmma<<<grid, dim3(128), 0, stream>>>(buf_agg, WL[L], buf_h, WR[L],
                                                       BLb[L], target, N, mstore);
        if (L < 2) {
            zero_f32<<<(1024 + 255) / 256, b256, 0, stream>>>((float*)sum, 1024);
            bn_stats<<<(N_NODES + BN_ROWS - 1) / BN_ROWS, b256, 0, stream>>>(buf_tmp, sum, ssq);
            bn_finalize<<<1, b256, 0, stream>>>(sum, ssq, mu, rs);
            bn_apply_relu<<<N_NODES, b256, 0, stream>>>(buf_tmp, mu, rs, G[L], BE[L], buf_h);
        }
    }
}